// LICv1_74448963109133
// MI455X (gfx1250) — compile-verified
//
#include <hip/hip_runtime.h>
#include <hip/hip_bf16.h>
#include <math.h>

// CLOCK-attention scorer for MI455X (gfx1250, wave32).
// Identity used: s = u.q = sum_l attn_l * (k_l . q) = softmax-weighted sum of sim.
// One wave per batch element; V_WMMA_F32_16X16X4_F32 scores 16 sequence rows
// against BOTH candidates (cols 0-7 = q_pos, cols 8-15 = q_neg) per WMMA chain.

typedef __attribute__((ext_vector_type(2))) float v2f;
typedef __attribute__((ext_vector_type(8))) float v8f;

#ifndef __has_builtin
#define __has_builtin(x) 0
#endif
#if __has_builtin(__builtin_amdgcn_wmma_f32_16x16x4_f32)
#define HAVE_WMMA_F32X4 1
#else
#define HAVE_WMMA_F32X4 0
#endif

#define WAVES_PER_BLOCK 8

__device__ __forceinline__ float shfl_f(float v, int src) { return __shfl(v, src, 32); }
__device__ __forceinline__ int   shfl_i(int v, int src)   { return __shfl(v, src, 32); }

// Returns softmax(sim*gate/tau) . sim for this lane's candidate column group.
// Lane layout: n = lane&15 selects candidate column (n<8 -> pos, else neg);
// hh = lane>>4 selects which 8 of the 16 tile rows this lane accumulates.
__device__ __forceinline__ float tower_score(
    const int* __restrict__ items, const int* __restrict__ dts,
    const unsigned char* __restrict__ mask,
    const float* __restrict__ emb, const float* __restrict__ gate_tab,
    float invtau, const float* qbuf /* LDS: [0..63]=q_pos, [64..127]=q_neg */,
    int b, int L, int lane)
{
    const int n  = lane & 15;
    const int hh = lane >> 4;
    const float* qsel = qbuf + ((n < 8) ? 0 : 64);

    float mx = -INFINITY, sum = 0.0f, wsum = 0.0f;
    const long base = (long)b * (long)L;

    for (int t0 = 0; t0 < L; t0 += 16) {
        // Per-row metadata: lane (and lane+16) own row t0+n.
        const int   it = items[base + t0 + n];
        const int   dt = dts  [base + t0 + n];
        const float gv = gate_tab[dt];
        const int   mk = (int)mask[base + t0 + n];

        v8f c = {0.f, 0.f, 0.f, 0.f, 0.f, 0.f, 0.f, 0.f};
#if HAVE_WMMA_F32X4
        {
            // A (16x4 f32): lane = 16*kh + m; a0=A[m][2kh], a1=A[m][2kh+1]
            const float* arow = emb + (long)it * 64 + 2 * hh;
            // B (4x16 f32): lane = 16*kh + n; b0=B[2kh][n], b1=B[2kh+1][n]
            const float* brow = qsel + 2 * hh;
            #pragma unroll
            for (int kc = 0; kc < 16; ++kc) {
                v2f a  = *(const v2f*)(arow + 4 * kc);
                v2f bm = *(const v2f*)(brow + 4 * kc);
                // D = A*B + C ; full fp32 precision (matches reference einsum)
                c = __builtin_amdgcn_wmma_f32_16x16x4_f32(
                        false, a, false, bm, (short)0, c, false, false);
            }
        }
#else
        {
            // Fallback: per-lane dot products with the same D-layout semantics.
            #pragma unroll
            for (int r = 0; r < 8; ++r) {
                const int j   = r + 8 * hh;
                const int itj = shfl_i(it, j);
                const float* kr = emb + (long)itj * 64;
                float acc = 0.0f;
                for (int d = 0; d < 64; ++d) acc += kr[d] * qsel[d];
                c[r] = acc;
            }
        }
#endif
        // Online softmax-weighted accumulation of sim.
        #pragma unroll
        for (int r = 0; r < 8; ++r) {
            const int   j    = r + 8 * hh;        // row within tile
            const float g    = shfl_f(gv, j);
            const int   mj   = shfl_i(mk, j);
            const float simv = c[r];
            const float logit = mj ? (simv * g * invtau) : -INFINITY;
            const float nm = fmaxf(mx, logit);
            if (nm != -INFINITY) {
                const float corr = expf(mx - nm);     // exp(-inf)=0 on first hit
                const float e    = expf(logit - nm);
                sum  = sum  * corr + e;
                wsum = wsum * corr + e * simv;
                mx = nm;
            }
        }
    }

    // Merge the two row-halves (lanes l and l^16 share the same candidate).
    {
        const float omx   = __shfl_xor(mx,   16, 32);
        const float osum  = __shfl_xor(sum,  16, 32);
        const float owsum = __shfl_xor(wsum, 16, 32);
        const float nm = fmaxf(mx, omx);
        if (nm != -INFINITY) {
            const float c1 = expf(mx  - nm);
            const float c2 = expf(omx - nm);
            sum  = sum  * c1 + osum  * c2;
            wsum = wsum * c1 + owsum * c2;
        } else { sum = 0.0f; wsum = 0.0f; }
    }
    return (sum > 0.0f) ? (wsum / sum) : 0.0f;
}

__global__ void __launch_bounds__(32 * WAVES_PER_BLOCK)
clock_score_kernel(const int* __restrict__ items_long, const int* __restrict__ dts_long,
                   const unsigned char* __restrict__ mask_long,
                   const int* __restrict__ items_short, const int* __restrict__ dts_short,
                   const unsigned char* __restrict__ mask_short,
                   const int* __restrict__ pos_items, const int* __restrict__ neg_items,
                   const float* __restrict__ emb,
                   const float* __restrict__ gate_g, const float* __restrict__ gate_e,
                   const float* __restrict__ raw_tau_g, const float* __restrict__ raw_tau_e,
                   const float* __restrict__ fuse_logits,
                   float* __restrict__ out, int B, int Llong, int Lshort)
{
    __shared__ __align__(16) float qbuf_all[WAVES_PER_BLOCK][128];
    const int lane = threadIdx.x & 31;
    const int wv   = threadIdx.x >> 5;
    const int b    = blockIdx.x * WAVES_PER_BLOCK + wv;
    float* qbuf = &qbuf_all[wv][0];

    if (b < B) {
        const long ip = (long)pos_items[b] * 64;
        const long iq = (long)neg_items[b] * 64;
        qbuf[lane]      = emb[ip + lane];
        qbuf[lane + 32] = emb[ip + lane + 32];
        qbuf[64 + lane] = emb[iq + lane];
        qbuf[96 + lane] = emb[iq + lane + 32];
    }
    __syncthreads();
    if (b >= B) return;

    const float tau_g = log1pf(expf(raw_tau_g[0])) + 1e-6f;   // softplus + eps
    const float tau_e = log1pf(expf(raw_tau_e[0])) + 1e-6f;
    const float lam   = 1.0f / (1.0f + expf(fuse_logits[1] - fuse_logits[0]));

    const float Sg = tower_score(items_long, dts_long, mask_long, emb, gate_g,
                                 1.0f / tau_g, qbuf, b, Llong, lane);
    const float Se = tower_score(items_short, dts_short, mask_short, emb, gate_e,
                                 1.0f / tau_e, qbuf, b, Lshort, lane);
    const float s = lam * Sg + (1.0f - lam) * Se;

    if (lane == 0) out[b] = s;          // lane 0: candidate column = pos
    if (lane == 8) out[B + b] = s;      // lane 8: candidate column = neg
    if (b == 0 && lane == 0) {
        out[2 * B + 0] = lam;
        out[2 * B + 1] = tau_g;
        out[2 * B + 2] = tau_e;
    }
}

extern "C" void kernel_launch(void* const* d_in, const int* in_sizes, int n_in,
                              void* d_out, int out_size, void* d_ws, size_t ws_size,
                              hipStream_t stream) {
    const int* items_long  = (const int*)d_in[0];
    const int* dts_long    = (const int*)d_in[1];
    const unsigned char* mask_long  = (const unsigned char*)d_in[2];
    const int* items_short = (const int*)d_in[3];
    const int* dts_short   = (const int*)d_in[4];
    const unsigned char* mask_short = (const unsigned char*)d_in[5];
    const int* pos_items   = (const int*)d_in[6];
    const int* neg_items   = (const int*)d_in[7];
    const float* emb       = (const float*)d_in[8];
    const float* gate_g    = (const float*)d_in[9];
    const float* gate_e    = (const float*)d_in[10];
    const float* raw_tau_g = (const float*)d_in[11];
    const float* raw_tau_e = (const float*)d_in[12];
    const float* fuse      = (const float*)d_in[13];

    const int B      = in_sizes[6];
    const int Llong  = in_sizes[0] / B;
    const int Lshort = in_sizes[3] / B;

    const int blocks = (B + WAVES_PER_BLOCK - 1) / WAVES_PER_BLOCK;
    clock_score_kernel<<<blocks, 32 * WAVES_PER_BLOCK, 0, stream>>>(
        items_long, dts_long, mask_long,
        items_short, dts_short, mask_short,
        pos_items, neg_items, emb, gate_g, gate_e,
        raw_tau_g, raw_tau_e, fuse,
        (float*)d_out, B, Llong, Lshort);
}